// AutoEnKF_40905268527432
// MI455X (gfx1250) — compile-verified
//
#include <hip/hip_runtime.h>
#include <math.h>

#define DXX 32
#define DYY 16
#define SEQL 128
#define NMC 50
#define NMCP 64          // padded ensemble rows (zeros beyond 50)
#define BATCH 32
#define NTHREADS 128
#define NWAVES 4

#define QVAR 0.0025f     // 0.05^2
#define RVAR 0.01f       // 0.1^2
#define RCHOL 0.1f
#define LOG2PI 1.8378770664093453f

typedef __attribute__((ext_vector_type(2))) float v2f;
typedef __attribute__((ext_vector_type(8))) float v8f;

// ---- shared memory pool layout (floats) ----
#define OFF_W1    0        // 32x128
#define OFF_B1    4096     // 128
#define OFF_W2    4224     // 128x64
#define OFF_B2    12416    // 64
#define OFF_W3    12480    // 64x32
#define OFF_B3    14528    // 32
#define OFF_W4    14560    // 32x32
#define OFF_B4    15584    // 32
#define OFF_TAP   15616    // 32x32 taper
#define OFF_X     16640    // 64x32 ensemble carry
#define OFF_H1    18688    // 64x128 (scratch region, aliased below)
#define OFF_H3    (OFF_H1 + 0)     // 64x32
#define OFF_XT    (OFF_H1 + 2048)  // 64x32 predicted ensemble
#define OFF_XC    (OFF_H1 + 4096)  // 64x32 centered
#define OFF_D     (OFF_H1 + 6144)  // 64x16 innovation rows
#define OFF_H2    26880    // 64x64 (aliased after GEMM3)
#define OFF_P     (OFF_H2 + 0)     // 32x32
#define OFF_L     (OFF_H2 + 1024)  // 16x16 cholesky (in place on S)
#define OFF_K     (OFF_H2 + 1280)  // 16x32 Kalman gain
#define OFF_XM    (OFF_H2 + 1792)  // 32
#define OFF_Y     (OFF_H2 + 1824)  // 16
#define OFF_YP    (OFF_H2 + 1840)  // 16
#define SMEM_FLOATS 30976          // ~121 KB << 320 KB/WGP

// ---- WMMA f32 16x16x4 fragment helpers (wave32) ----
// A 16x4: lane<16 -> row=lane, K={4k,4k+1}; lane>=16 -> row=lane-16, K={4k+2,4k+3}
__device__ __forceinline__ v2f lda_frag(const float* A, int lda, int m0, int k0, int lane) {
    int r  = lane & 15;
    int kk = k0 + ((lane >> 4) << 1);
    const float* p = A + (m0 + r) * lda + kk;
    v2f a; a.x = p[0]; a.y = p[1]; return a;
}
// A = X^T: A[m][k] = X[k][m]
__device__ __forceinline__ v2f ldat_frag(const float* X, int ldx, int m0, int k0, int lane) {
    int r  = lane & 15;
    int kk = k0 + ((lane >> 4) << 1);
    v2f a; a.x = X[kk * ldx + m0 + r]; a.y = X[(kk + 1) * ldx + m0 + r]; return a;
}
// B 4x16 (row-major KxN): lane<16 -> col=lane, K={4k,4k+1}; lane>=16 -> col=lane-16, K={4k+2,4k+3}
__device__ __forceinline__ v2f ldb_frag(const float* B, int ldb, int k0, int n0, int lane) {
    int c  = lane & 15;
    int kk = k0 + ((lane >> 4) << 1);
    v2f b; b.x = B[kk * ldb + n0 + c]; b.y = B[(kk + 1) * ldb + n0 + c]; return b;
}
// C/D 16x16: vgpr r: lane<16 -> (m0+r, n0+lane); lane>=16 -> (m0+8+r, n0+lane-16)
__device__ __forceinline__ v8f ldc_frag(const float* C, int ldc, int m0, int n0, int lane) {
    int c  = n0 + (lane & 15);
    int mb = m0 + ((lane >> 4) << 3);
    v8f v;
#pragma unroll
    for (int r = 0; r < 8; ++r) v[r] = C[(mb + r) * ldc + c];
    return v;
}
__device__ __forceinline__ void stc_frag(float* C, int ldc, int m0, int n0, int lane, v8f v) {
    int c  = n0 + (lane & 15);
    int mb = m0 + ((lane >> 4) << 3);
#pragma unroll
    for (int r = 0; r < 8; ++r) C[(mb + r) * ldc + c] = v[r];
}

// Generic LDS GEMM: C = op(A[MxK] * B[KxN] + bias (+ Cin)), tiles striped over waves.
__device__ __forceinline__ void gemm_wmma(const float* A, int lda, const float* B, int ldb,
                                          float* C, int ldc, const float* bias,
                                          const float* Cin, int M, int N, int K,
                                          bool relu, int tid) {
    int lane = tid & 31, wave = tid >> 5;
    int mt = M >> 4, nt = N >> 4, kt = K >> 2;
    for (int t = wave; t < mt * nt; t += NWAVES) {
        int i = t / nt, j = t % nt;
        v8f acc;
        if (Cin) acc = ldc_frag(Cin, ldc, i * 16, j * 16, lane);
        else { v8f z = {}; acc = z; }
        for (int k = 0; k < kt; ++k) {
            v2f a = lda_frag(A, lda, i * 16, k * 4, lane);
            v2f b = ldb_frag(B, ldb, k * 4, j * 16, lane);
            acc = __builtin_amdgcn_wmma_f32_16x16x4_f32(false, a, false, b,
                                                        (short)0, acc, false, false);
        }
        float bv = bias ? bias[j * 16 + (lane & 15)] : 0.0f;
#pragma unroll
        for (int r = 0; r < 8; ++r) {
            float v = acc[r] + bv;
            if (relu) v = fmaxf(v, 0.0f);
            acc[r] = v;
        }
        stc_frag(C, ldc, i * 16, j * 16, lane, acc);
    }
}

__global__ __launch_bounds__(NTHREADS) void enkf_persistent(
    const float* __restrict__ obs,    // [B,SEQ,DY]
    const float* __restrict__ x0,     // [N,B,DX]
    const float* __restrict__ noise,  // [SEQ,B,DY]
    const float* __restrict__ gW1, const float* __restrict__ gb1,
    const float* __restrict__ gW2, const float* __restrict__ gb2,
    const float* __restrict__ gW3, const float* __restrict__ gb3,
    const float* __restrict__ gW4, const float* __restrict__ gb4,
    float* __restrict__ ll_ws,        // [B] per-block ll sums
    float* __restrict__ out_mean,     // [B,SEQ,DX]
    float* __restrict__ out_var)      // [B,SEQ,DX]
{
    __shared__ float sm[SMEM_FLOATS];
    const int tid = threadIdx.x;
    const int b   = blockIdx.x;

    // ---- load weights into LDS (reused for all 128 steps) ----
    for (int i = tid; i < 32 * 128; i += NTHREADS) sm[OFF_W1 + i] = gW1[i];
    for (int i = tid; i < 128;      i += NTHREADS) sm[OFF_B1 + i] = gb1[i];
    for (int i = tid; i < 128 * 64; i += NTHREADS) sm[OFF_W2 + i] = gW2[i];
    for (int i = tid; i < 64;       i += NTHREADS) sm[OFF_B2 + i] = gb2[i];
    for (int i = tid; i < 64 * 32;  i += NTHREADS) sm[OFF_W3 + i] = gW3[i];
    for (int i = tid; i < 32;       i += NTHREADS) sm[OFF_B3 + i] = gb3[i];
    for (int i = tid; i < 32 * 32;  i += NTHREADS) sm[OFF_W4 + i] = gW4[i];
    for (int i = tid; i < 32;       i += NTHREADS) sm[OFF_B4 + i] = gb4[i];

    // ---- Gaspari-Cohn taper (fixed 32x32) ----
    for (int idx = tid; idx < 32 * 32; idx += NTHREADS) {
        int i = idx >> 5, j = idx & 31;
        int diff = i - j; if (diff < 0) diff = -diff;
        int dd = (diff < 32 - diff) ? diff : 32 - diff;
        float z  = (float)dd * (1.0f / 5.0f);
        float z2 = z * z, z3 = z2 * z, z4 = z2 * z2, z5 = z4 * z;
        float t;
        if (z < 1.0f) {
            t = 1.0f - (5.0f/3.0f)*z2 + (5.0f/8.0f)*z3 + 0.5f*z4 - 0.25f*z5;
        } else if (z < 2.0f) {
            float zs = fmaxf(z, 1e-12f);
            t = 4.0f - 5.0f*z + (5.0f/3.0f)*z2 + (5.0f/8.0f)*z3 - 0.5f*z4
                + (1.0f/12.0f)*z5 - 2.0f/(3.0f*zs);
        } else t = 0.0f;
        sm[OFF_TAP + idx] = t;
    }

    // ---- initial ensemble (rows 50..63 zero-padded) ----
    for (int i = tid; i < NMCP * DXX; i += NTHREADS) {
        int n = i >> 5, j = i & 31;
        sm[OFF_X + i] = (n < NMC) ? x0[(n * BATCH + b) * DXX + j] : 0.0f;
    }
    __syncthreads();

    float ll_acc = 0.0f;   // only meaningful on tid==0

    for (int t = 0; t < SEQL; ++t) {
        // ---- MLP prediction: 4 WMMA GEMMs over the 64x32 (padded) ensemble ----
        gemm_wmma(sm + OFF_X,  DXX, sm + OFF_W1, 128, sm + OFF_H1, 128,
                  sm + OFF_B1, nullptr, NMCP, 128, 32, true, tid);
        __syncthreads();
        gemm_wmma(sm + OFF_H1, 128, sm + OFF_W2, 64,  sm + OFF_H2, 64,
                  sm + OFF_B2, nullptr, NMCP, 64, 128, true, tid);
        __syncthreads();
        gemm_wmma(sm + OFF_H2, 64,  sm + OFF_W3, 32,  sm + OFF_H3, 32,
                  sm + OFF_B3, nullptr, NMCP, 32, 64, true, tid);
        __syncthreads();
        gemm_wmma(sm + OFF_H3, 32,  sm + OFF_W4, 32,  sm + OFF_XT, 32,
                  sm + OFF_B4, nullptr, NMCP, 32, 32, false, tid);
        __syncthreads();

        // ---- ensemble mean + load y / perturbed y ----
        if (tid < DXX) {
            float s = 0.0f;
            for (int n = 0; n < NMC; ++n) s += sm[OFF_XT + n * DXX + tid];
            sm[OFF_XM + tid] = s * (1.0f / NMC);
        }
        if (tid >= 64 && tid < 64 + DYY) {
            int i = tid - 64;
            float y = obs[(b * SEQL + t) * DYY + i];
            float e = noise[(t * BATCH + b) * DYY + i];
            sm[OFF_Y  + i] = y;
            sm[OFF_YP + i] = y + e * RCHOL;
        }
        __syncthreads();

        // ---- center (masked so padded rows contribute 0 to P) ----
        for (int i = tid; i < NMCP * DXX; i += NTHREADS) {
            int n = i >> 5, j = i & 31;
            sm[OFF_XC + i] = (n < NMC) ? (sm[OFF_XT + i] - sm[OFF_XM + j]) : 0.0f;
        }
        __syncthreads();

        // ---- P = Xc^T Xc via WMMA (A transposed-load), 2x2 tiles, K=64 ----
        {
            int lane = tid & 31, wave = tid >> 5;
            for (int tt = wave; tt < 4; tt += NWAVES) {
                int i = tt >> 1, j = tt & 1;
                v8f acc = {};
                for (int k = 0; k < 16; ++k) {
                    v2f a = ldat_frag(sm + OFF_XC, DXX, i * 16, k * 4, lane);
                    v2f bb = ldb_frag(sm + OFF_XC, DXX, k * 4, j * 16, lane);
                    acc = __builtin_amdgcn_wmma_f32_16x16x4_f32(false, a, false, bb,
                                                                (short)0, acc, false, false);
                }
                stc_frag(sm + OFF_P, DXX, i * 16, j * 16, lane, acc);
            }
        }
        __syncthreads();

        // ---- P = taper .* P/(N-1) + Q ----
        for (int idx = tid; idx < 32 * 32; idx += NTHREADS) {
            int i = idx >> 5, j = idx & 31;
            float p = sm[OFF_P + idx] * (1.0f / (NMC - 1)) * sm[OFF_TAP + idx];
            if (i == j) p += QVAR;
            sm[OFF_P + idx] = p;
        }
        __syncthreads();

        // ---- S = P[:16,:16] + R (into L, factored in place); innovation rows D ----
        for (int idx = tid; idx < 256; idx += NTHREADS) {
            int i = idx >> 4, j = idx & 15;
            sm[OFF_L + idx] = sm[OFF_P + i * DXX + j] + ((i == j) ? RVAR : 0.0f);
        }
        for (int idx = tid; idx < NMCP * DYY; idx += NTHREADS) {
            int n = idx >> 4, i = idx & 15;
            sm[OFF_D + idx] = (n < NMC) ? (sm[OFF_YP + i] - sm[OFF_XT + n * DXX + i]) : 0.0f;
        }
        __syncthreads();

        // ---- Cholesky of S (16x16, barriered column sweep) ----
        for (int k = 0; k < DYY; ++k) {
            if (tid == 0) {
                float s = sm[OFF_L + k * 16 + k];
                for (int j = 0; j < k; ++j) { float v = sm[OFF_L + k * 16 + j]; s -= v * v; }
                sm[OFF_L + k * 16 + k] = sqrtf(s);
            }
            __syncthreads();
            if (tid > k && tid < DYY) {
                float s = sm[OFF_L + tid * 16 + k];
                for (int j = 0; j < k; ++j)
                    s -= sm[OFF_L + tid * 16 + j] * sm[OFF_L + k * 16 + j];
                sm[OFF_L + tid * 16 + k] = s / sm[OFF_L + k * 16 + k];
            }
            __syncthreads();
        }

        // ---- K = S^-1 HP : fwd+back substitution, one thread per column ----
        if (tid < DXX) {
            float z[DYY];
            for (int i = 0; i < DYY; ++i) {
                float s = sm[OFF_P + i * DXX + tid];          // HP[i][c]
                for (int j = 0; j < i; ++j) s -= sm[OFF_L + i * 16 + j] * z[j];
                z[i] = s / sm[OFF_L + i * 16 + i];
            }
            for (int i = DYY - 1; i >= 0; --i) {
                float s = z[i];
                for (int j = i + 1; j < DYY; ++j) s -= sm[OFF_L + j * 16 + i] * z[j];
                z[i] = s / sm[OFF_L + i * 16 + i];
            }
            for (int i = 0; i < DYY; ++i) sm[OFF_K + i * DXX + tid] = z[i];
        }
        __syncthreads();

        // ---- ensemble update: X = XT + D @ K (WMMA, C initialized from XT) ----
        gemm_wmma(sm + OFF_D, DYY, sm + OFF_K, DXX, sm + OFF_X, DXX,
                  nullptr, sm + OFF_XT, NMCP, DXX, DYY, false, tid);

        // ---- mean update, posterior variance diag (outputs) ----
        if (tid < DXX) {
            float s = 0.0f, pv = sm[OFF_P + tid * DXX + tid];
            for (int i = 0; i < DYY; ++i) {
                float ki = sm[OFF_K + i * DXX + tid];
                s  += (sm[OFF_Y + i] - sm[OFF_XM + i]) * ki;
                pv -= sm[OFF_P + i * DXX + tid] * ki;
            }
            int o = (b * SEQL + t) * DXX + tid;
            out_mean[o] = sm[OFF_XM + tid] + s;
            out_var[o]  = pv;
        }
        // ---- log-likelihood N(y | H xm, S) ----
        if (tid == 0) {
            float zz[DYY];
            float ll = -0.5f * (float)DYY * LOG2PI;
            for (int i = 0; i < DYY; ++i) {
                float s = sm[OFF_Y + i] - sm[OFF_XM + i];
                for (int j = 0; j < i; ++j) s -= sm[OFF_L + i * 16 + j] * zz[j];
                float d = sm[OFF_L + i * 16 + i];
                zz[i] = s / d;
                ll -= 0.5f * zz[i] * zz[i];
                ll -= logf(d);
            }
            ll_acc += ll;
        }
        __syncthreads();
    }

    if (tid == 0) ll_ws[b] = ll_acc;
}

__global__ void enkf_reduce_ll(const float* __restrict__ ll_ws, float* __restrict__ out) {
    if (threadIdx.x == 0 && blockIdx.x == 0) {
        float s = 0.0f;
        for (int i = 0; i < BATCH; ++i) s += ll_ws[i];   // fixed order: deterministic
        out[0] = s * (1.0f / BATCH);
    }
}

extern "C" void kernel_launch(void* const* d_in, const int* in_sizes, int n_in,
                              void* d_out, int out_size, void* d_ws, size_t ws_size,
                              hipStream_t stream) {
    (void)in_sizes; (void)n_in; (void)out_size; (void)ws_size;
    const float* obs   = (const float*)d_in[0];
    const float* x0    = (const float*)d_in[1];
    const float* noise = (const float*)d_in[2];
    const float* W1 = (const float*)d_in[3];  const float* b1 = (const float*)d_in[4];
    const float* W2 = (const float*)d_in[5];  const float* b2 = (const float*)d_in[6];
    const float* W3 = (const float*)d_in[7];  const float* b3 = (const float*)d_in[8];
    const float* W4 = (const float*)d_in[9];  const float* b4 = (const float*)d_in[10];

    float* out = (float*)d_out;
    float* out_mean = out + 1;
    float* out_var  = out + 1 + BATCH * SEQL * DXX;
    float* ll_ws    = (float*)d_ws;

    enkf_persistent<<<BATCH, NTHREADS, 0, stream>>>(
        obs, x0, noise, W1, b1, W2, b2, W3, b3, W4, b4,
        ll_ws, out_mean, out_var);
    enkf_reduce_ll<<<1, 32, 0, stream>>>(ll_ws, out);
}